// GaussianRenderer_33320356282417
// MI455X (gfx1250) — compile-verified
//
#include <hip/hip_runtime.h>
#include <hip/hip_bf16.h>

#define HW_IN   (512 * 512)        // per-batch gaussians
#define W_OUT   1024
#define HW_OUT  (1024 * 1024)
#define NBATCH  4
#define NPLANES 8                  // uv c0,c1,c10,c11,c12,c13 + pos c0,c1

// ---------------------------------------------------------------- zero init
__global__ __launch_bounds__(256) void gr_zero(float4* __restrict__ out, int n4) {
    int i = blockIdx.x * 256 + threadIdx.x;
    if (i < n4) out[i] = make_float4(0.f, 0.f, 0.f, 0.f);
}

// ---------------------------------------------------------------- scatter
__global__ __launch_bounds__(256) void gr_scatter(const float* __restrict__ uv,
                                                  const float* __restrict__ pos,
                                                  float* __restrict__ out) {
    __shared__ float stage[NPLANES * 256];

    const unsigned tid = threadIdx.x;
    const unsigned g   = blockIdx.x * 256u + tid;       // global gaussian id
    const unsigned b   = g / HW_IN;                     // batch (uniform per block)
    const unsigned s   = g - b * HW_IN;                 // spatial index in 512x512

    const float* up = uv  + (size_t)b * 14 * HW_IN;     // uv batch base
    const float* pp = pos + (size_t)b * 3  * HW_IN;     // position batch base

    float u0, u1, u10, u11, u12, u13, p0, p1;

#if __has_builtin(__builtin_amdgcn_global_load_async_to_lds_b32)
    // Stage the 8 channel planes through LDS with the gfx1250 async DMA engine.
    // Prototype on this toolchain: (int* global_src, int* lds_dst, imm offset, imm cpol)
    __builtin_amdgcn_global_load_async_to_lds_b32((int*)(up + (size_t)0  * HW_IN + s), (int*)&stage[0 * 256 + tid], 0, 0);
    __builtin_amdgcn_global_load_async_to_lds_b32((int*)(up + (size_t)1  * HW_IN + s), (int*)&stage[1 * 256 + tid], 0, 0);
    __builtin_amdgcn_global_load_async_to_lds_b32((int*)(up + (size_t)10 * HW_IN + s), (int*)&stage[2 * 256 + tid], 0, 0);
    __builtin_amdgcn_global_load_async_to_lds_b32((int*)(up + (size_t)11 * HW_IN + s), (int*)&stage[3 * 256 + tid], 0, 0);
    __builtin_amdgcn_global_load_async_to_lds_b32((int*)(up + (size_t)12 * HW_IN + s), (int*)&stage[4 * 256 + tid], 0, 0);
    __builtin_amdgcn_global_load_async_to_lds_b32((int*)(up + (size_t)13 * HW_IN + s), (int*)&stage[5 * 256 + tid], 0, 0);
    __builtin_amdgcn_global_load_async_to_lds_b32((int*)(pp + (size_t)0  * HW_IN + s), (int*)&stage[6 * 256 + tid], 0, 0);
    __builtin_amdgcn_global_load_async_to_lds_b32((int*)(pp + (size_t)1  * HW_IN + s), (int*)&stage[7 * 256 + tid], 0, 0);
#if __has_builtin(__builtin_amdgcn_s_wait_asynccnt)
    __builtin_amdgcn_s_wait_asynccnt(0);
#else
    asm volatile("s_wait_asynccnt 0" ::: "memory");
#endif
    // Each lane reads back exactly the slots it requested -> wave-level wait
    // is sufficient; no workgroup barrier required.
    u0  = stage[0 * 256 + tid];
    u1  = stage[1 * 256 + tid];
    u10 = stage[2 * 256 + tid];
    u11 = stage[3 * 256 + tid];
    u12 = stage[4 * 256 + tid];
    u13 = stage[5 * 256 + tid];
    p0  = stage[6 * 256 + tid];
    p1  = stage[7 * 256 + tid];
#else
    u0  = up[(size_t)0  * HW_IN + s];
    u1  = up[(size_t)1  * HW_IN + s];
    u10 = up[(size_t)10 * HW_IN + s];
    u11 = up[(size_t)11 * HW_IN + s];
    u12 = up[(size_t)12 * HW_IN + s];
    u13 = up[(size_t)13 * HW_IN + s];
    p0  = pp[(size_t)0  * HW_IN + s];
    p1  = pp[(size_t)1  * HW_IN + s];
#endif

    // positions (x,y only; z never used downstream)
    const float posx = p0 + u0;
    const float posy = p1 + u1;

    // opacity = sigmoid(ch10); vals = colors * opacity
    const float op = 1.0f / (1.0f + __expf(-u10));
    const float vr = u11 * op;
    const float vg = u12 * op;
    const float vb = u13 * op;

    // NDC -> pixel: truncating int cast (matches .astype(int32)), then clamp
    int px = (int)((posx + 1.0f) * 512.0f);   // 0.5 * 1024
    int py = (int)((posy + 1.0f) * 512.0f);
    px = min(max(px, 0), W_OUT - 1);
    py = min(max(py, 0), W_OUT - 1);

    const size_t base = (size_t)b * 3 * HW_OUT + (size_t)py * W_OUT + (size_t)px;
    atomicAdd(&out[base + (size_t)0 * HW_OUT], vr);
    atomicAdd(&out[base + (size_t)1 * HW_OUT], vg);
    atomicAdd(&out[base + (size_t)2 * HW_OUT], vb);
}

// ---------------------------------------------------------------- clamp
__global__ __launch_bounds__(256) void gr_clamp(float4* __restrict__ out, int n4) {
    int i = blockIdx.x * 256 + threadIdx.x;
    if (i < n4) {
        float4 v = out[i];
        v.x = fminf(fmaxf(v.x, 0.f), 1.f);
        v.y = fminf(fmaxf(v.y, 0.f), 1.f);
        v.z = fminf(fmaxf(v.z, 0.f), 1.f);
        v.w = fminf(fmaxf(v.w, 0.f), 1.f);
        out[i] = v;
    }
}

extern "C" void kernel_launch(void* const* d_in, const int* in_sizes, int n_in,
                              void* d_out, int out_size, void* d_ws, size_t ws_size,
                              hipStream_t stream) {
    const float* uv  = (const float*)d_in[0];   // [4,14,512,512]
    const float* pos = (const float*)d_in[1];   // [4,3,512,512]
    float*       out = (float*)d_out;           // [4,3,1024,1024]

    const int n_elems = NBATCH * 3 * HW_OUT;    // 12,582,912
    const int n4      = n_elems / 4;            // 3,145,728 (exact)

    // 1) framebuffer must restart at zero every call (graph replays reuse d_out)
    gr_zero<<<n4 / 256, 256, 0, stream>>>((float4*)out, n4);

    // 2) scatter-add: one thread per gaussian, 4096 blocks x 256 threads (exact)
    gr_scatter<<<(NBATCH * HW_IN) / 256, 256, 0, stream>>>(uv, pos, out);

    // 3) clip to [0,1] in place
    gr_clamp<<<n4 / 256, 256, 0, stream>>>((float4*)out, n4);
}